// CholeskyBenchmarkModel_3083786518607
// MI455X (gfx1250) — compile-verified
//
#include <hip/hip_runtime.h>
#include <math.h>

typedef float v2f __attribute__((ext_vector_type(2)));
typedef float v8f __attribute__((ext_vector_type(8)));
typedef unsigned int v4u __attribute__((ext_vector_type(4)));
typedef int v8i __attribute__((ext_vector_type(8)));
typedef int v4i __attribute__((ext_vector_type(4)));

#define BQ   65536
#define NQ   40
#define NN   (NQ * NQ)          // 1600
#define LDR  49                 // padded LDS row stride (odd -> conflict-free)
#define PADN 48                 // padded matrix dim (3 x 16 WMMA tiles)
#define SLICE (PADN * LDR)      // 2352 floats per matrix slice
#define WPB   4                 // waves per block
#define BTH   (WPB * 32)        // 128 threads
#define NBLK  4096              // 4096 blocks * 4 waves = 16384 waves -> 4 matrices/wave

// TDM copy: one 40-float row per iteration, 40 iterations,
// global advances 40 elements/row, LDS advances 49 elements/row (padded stride).
__device__ __forceinline__ void tdm_load_tile(const float* gsrc, unsigned lds_byte_off)
{
#if __has_builtin(__builtin_amdgcn_tensor_load_to_lds)
    unsigned long long ga = (unsigned long long)(uintptr_t)gsrc;   // byte address
    v4u g0;
    g0.x = 1u;                                  // count=1, gather off, user mode
    g0.y = lds_byte_off;                        // lds_addr (bytes)
    g0.z = (unsigned)(ga & 0xFFFFFFFFu);        // global_addr[31:0]
    g0.w = (unsigned)((ga >> 32) & 0x1FFFFFFu)  // global_addr[56:32]
         | 0x80000000u;                         // type=2 ("image") in bits 127:126
    v8i g1;
    g1[0] = (2 << 16) | (1 << 19);              // data_size=4B, iterate_enable=1
    g1[1] = (NQ & 0xFFFF) << 16;                // tensor_dim0[15:0]
    g1[2] = (NQ & 0xFFFF) << 16;                // tensor_dim0[31:16]=0 | tensor_dim1[15:0]
    g1[3] = (NQ & 0xFFFF) << 16;                // tensor_dim1[31:16]=0 | tile_dim0=40
    g1[4] = 1;                                  // tile_dim1=1, tile_dim2=0
    g1[5] = NQ;                                 // tensor_dim0_stride[31:0] = 40
    g1[6] = (NN & 0xFFFF) << 16;                // dim0_stride[47:32]=0 | dim1_stride[15:0]
    g1[7] = 0;                                  // dim1_stride[47:16] = 0
    v4i g2;
    g2[0] = 1;                                  // tensor_dim2 (unused; tile_dim2=0)
    g2[1] = LDR;                                // lds_addr_increment (data_size units)
    g2[2] = NQ;                                 // global_addr_increment[31:0]
    g2[3] = (NQ - 1) << 16;                     // incr[47:32]=0 | iterate_count=39 -> 40x
    v4i g3 = {0, 0, 0, 0};
    v8i g4 = {0, 0, 0, 0, 0, 0, 0, 0};          // extra group (6-arg toolchain variant)
    __builtin_amdgcn_tensor_load_to_lds(g0, g1, g2, g3, g4, 0);
    __builtin_amdgcn_s_wait_tensorcnt(0);
#else
    (void)gsrc; (void)lds_byte_off;
#endif
}

__global__ __launch_bounds__(BTH)
void chol_main(const float* __restrict__ Aall,
               const float* __restrict__ Refall,
               float* __restrict__ outL,
               float* __restrict__ outRef,
               float* __restrict__ partials)
{
    __shared__ float lds[WPB * SLICE];
    __shared__ float red[WPB * 3];

    const int tid    = threadIdx.x;
    const int wave   = tid >> 5;
    const int lane   = tid & 31;
    const int lane16 = lane & 15;
    const int hi     = lane >> 4;          // 0: lanes 0-15, 1: lanes 16-31
    float* Ls = &lds[wave * SLICE];
    const unsigned ls_off = (unsigned)(uintptr_t)Ls;   // LDS byte offset (low 32 bits)

    // Zero the whole padded slice once; the 40x40 live region is rewritten per
    // matrix, the pad (rows/cols 40..47) stays zero forever.
    for (int i = lane; i < SLICE; i += 32) Ls[i] = 0.0f;
    __syncthreads();

    const int totalWaves = NBLK * WPB;
    const int waveG      = blockIdx.x * WPB + wave;

    float sumsq  = 0.0f;
    float maxfe  = 0.0f;
    float maxres = 0.0f;

    for (int b = waveG; b < BQ; b += totalWaves) {
        const float* A  = Aall   + (size_t)b * NN;
        const float* R  = Refall + (size_t)b * NN;
        float*       Lo = outL   + (size_t)b * NN;
        float*       Ro = outRef + (size_t)b * NN;

        // Prefetch next matrix (global_prefetch_b8)
        if (b + totalWaves < BQ)
            __builtin_prefetch(Aall + (size_t)(b + totalWaves) * NN + lane * 16, 0, 3);

        // ---- load A into LDS (40x40 region, padded stride 49) ----
#if __has_builtin(__builtin_amdgcn_tensor_load_to_lds)
        tdm_load_tile(A, ls_off);        // Tensor Data Mover, per-wave DMA
#else
        for (int i = lane; i < NN; i += 32) {
            int r = i / NQ, c = i - r * NQ;
            Ls[r * LDR + c] = A[i];
        }
#endif
        __syncthreads();

        // ---- right-looking Cholesky, one wave per matrix ----
        for (int j = 0; j < NQ; ++j) {
            float s   = sqrtf(Ls[j * LDR + j]);   // broadcast LDS read
            float inv = 1.0f / s;
            if (lane == 0) Ls[j * LDR + j] = s;
            for (int r = j + 1 + lane; r < NQ; r += 32)
                Ls[r * LDR + j] *= inv;
            __syncthreads();
            for (int r = j + 1 + lane; r < NQ; r += 32) {
                float lrj = Ls[r * LDR + j];
                for (int k = j + 1; k <= r; ++k)
                    Ls[r * LDR + k] -= lrj * Ls[k * LDR + j];
            }
            __syncthreads();
        }

        // ---- fused output pass: write L (upper tri = 0), copy reference,
        //      accumulate factor-error stats, zero upper triangle in LDS ----
        for (int i = lane; i < NN; i += 32) {
            int r = i / NQ, c = i - r * NQ;
            float v;
            if (c <= r) {
                v = Ls[r * LDR + c];
            } else {
                v = 0.0f;
                Ls[r * LDR + c] = 0.0f;   // make LDS hold exact lower-tri L
            }
            Lo[i] = v;
            float rf = R[i];
            Ro[i] = rf;
            float d = v - rf;
            sumsq += d * d;
            maxfe  = fmaxf(maxfe, fabsf(d));
        }
        __syncthreads();

        // ---- residual check via WMMA: recon = L * L^T on padded 48x48.
        //      Residual is symmetric -> lower-triangle tiles suffice. ----
        for (int bi = 0; bi < 3; ++bi) {
            for (int bj = 0; bj <= bi; ++bj) {
                v8f acc = {};
                const int arow = (bi * 16 + lane16) * LDR;
                const int brow = (bj * 16 + lane16) * LDR;
                for (int k0 = 0; k0 < PADN; k0 += 4) {
                    // A-frag 16x4 f32: VGPR0 = K {0 | 2}, VGPR1 = K {1 | 3}
                    // B-frag (= L^T slice) loads with the identical pattern.
                    int ka = k0 + (hi ? 2 : 0);
                    v2f af, bf;
                    af.x = Ls[arow + ka];
                    af.y = Ls[arow + ka + 1];
                    bf.x = Ls[brow + ka];
                    bf.y = Ls[brow + ka + 1];
                    acc = __builtin_amdgcn_wmma_f32_16x16x4_f32(
                        /*neg_a=*/false, af, /*neg_b=*/false, bf,
                        /*c_mod=*/(short)0, acc,
                        /*reuse_a=*/false, /*reuse_b=*/false);
                }
                // D layout: VGPR v -> lanes 0-15: (M=v, N=lane16),
                //                     lanes 16-31: (M=v+8, N=lane16)
                // Branchless compare: clamp address, mask the contribution.
#pragma unroll
                for (int v = 0; v < 8; ++v) {
                    int m = bi * 16 + v + (hi ? 8 : 0);
                    int n = bj * 16 + lane16;
                    bool inb = (m < NQ) & (n < NQ);
                    int mc = inb ? m : (NQ - 1);
                    int nc = inb ? n : (NQ - 1);
                    float aval = A[mc * NQ + nc];
                    float rres = fabsf(acc[v] - aval);
                    maxres = fmaxf(maxres, inb ? rres : 0.0f);
                }
            }
        }
        __syncthreads();   // before next matrix overwrites LDS
    }

    // ---- wave reduction (wave32) ----
    for (int off = 16; off > 0; off >>= 1) {
        sumsq += __shfl_down(sumsq, off, 32);
        maxfe  = fmaxf(maxfe,  __shfl_down(maxfe,  off, 32));
        maxres = fmaxf(maxres, __shfl_down(maxres, off, 32));
    }
    if (lane == 0) {
        red[wave * 3 + 0] = sumsq;
        red[wave * 3 + 1] = maxfe;
        red[wave * 3 + 2] = maxres;
    }
    __syncthreads();
    if (tid == 0) {
        float s = 0.0f, m1 = 0.0f, m2 = 0.0f;
        for (int w = 0; w < WPB; ++w) {
            s  += red[w * 3 + 0];
            m1  = fmaxf(m1, red[w * 3 + 1]);
            m2  = fmaxf(m2, red[w * 3 + 2]);
        }
        partials[(size_t)blockIdx.x * 4 + 0] = s;
        partials[(size_t)blockIdx.x * 4 + 1] = m1;
        partials[(size_t)blockIdx.x * 4 + 2] = m2;
        partials[(size_t)blockIdx.x * 4 + 3] = 0.0f;
    }
}

__global__ __launch_bounds__(256)
void chol_finalize(const float* __restrict__ partials, float* __restrict__ out, int nPart)
{
    __shared__ double ssum[256];
    __shared__ float  sm1[256];
    __shared__ float  sm2[256];
    int tid = threadIdx.x;
    double s = 0.0; float m1 = 0.0f, m2 = 0.0f;
    for (int i = tid; i < nPart; i += 256) {
        s  += (double)partials[(size_t)i * 4 + 0];
        m1  = fmaxf(m1, partials[(size_t)i * 4 + 1]);
        m2  = fmaxf(m2, partials[(size_t)i * 4 + 2]);
    }
    ssum[tid] = s; sm1[tid] = m1; sm2[tid] = m2;
    __syncthreads();
    for (int off = 128; off > 0; off >>= 1) {
        if (tid < off) {
            ssum[tid] += ssum[tid + off];
            sm1[tid]   = fmaxf(sm1[tid], sm1[tid + off]);
            sm2[tid]   = fmaxf(sm2[tid], sm2[tid + off]);
        }
        __syncthreads();
    }
    if (tid == 0) {
        out[0] = (float)(ssum[0] / ((double)BQ * (double)NN));   // loss
        size_t tail = (size_t)1 + 2 * (size_t)BQ * (size_t)NN;
        out[tail]     = sm1[0];   // max_factor_error
        out[tail + 1] = sm2[0];   // max_residual
    }
}

extern "C" void kernel_launch(void* const* d_in, const int* in_sizes, int n_in,
                              void* d_out, int out_size, void* d_ws, size_t ws_size,
                              hipStream_t stream) {
    const float* A   = (const float*)d_in[0];   // matrix   [B,N,N] fp32
    const float* Ref = (const float*)d_in[1];   // reference[B,N,N] fp32
    float* out    = (float*)d_out;
    float* outL   = out + 1;                              // tuple slot 1: L
    float* outRef = out + 1 + (size_t)BQ * (size_t)NN;    // tuple slot 2: reference copy
    float* partials = (float*)d_ws;                       // 4096 * 4 floats = 64 KB

    chol_main<<<NBLK, BTH, 0, stream>>>(A, Ref, outL, outRef, partials);
    chol_finalize<<<1, 256, 0, stream>>>(partials, out, NBLK);
}